// Net_78323023610515
// MI455X (gfx1250) — compile-verified
//
#include <hip/hip_runtime.h>
#include <hip/hip_bf16.h>

// ---------------------------------------------------------------------------
// Fused SPDNet forward for MI455X (gfx1250, wave32, WMMA + TDM).
// One wave per batch element:
//   X tile staged global->LDS via TENSOR_LOAD_TO_LDS (TDM, TENSORcnt)
//   stage1: Y1 = W1^T X W1 via V_WMMA_F32_16X16X4_F32
//   Jacobi eigh(20) -> rectify -> VALU transform (20->10)
//   Jacobi eigh(10) -> rectify -> VALU transform (10->2)
//   closed-form 2x2 rectify+log -> triu vec -> 17-wide linear head
// ---------------------------------------------------------------------------

typedef float v2f __attribute__((ext_vector_type(2)));
typedef float v8f __attribute__((ext_vector_type(8)));
typedef unsigned int v4u __attribute__((ext_vector_type(4)));
typedef int v4i __attribute__((ext_vector_type(4)));
typedef int v8i __attribute__((ext_vector_type(8)));

#define RECT_EPS 1e-4f
#define SQRT2_F  1.4142135623730951f

#define DD 57   // input dim (compact LDS stride)
#define NP 32   // N padded (20 -> 32, multiple of 16)

#define WPB     2          // waves per block
#define THREADS (WPB * 32)

// per-wave LDS arena (float offsets)
#define ARENA_F  6400
#define OFF_XPAD 0         // 57*57 = 3249 floats, compact (reused as H1/H2 later)
#define OFF_TPAD 3264      // 64*32 = 2048 floats (reused as T2/T3 later)
#define OFF_A20  5312      // 400
#define OFF_U20  5712      // 400
#define OFF_A10  6112      // 100
#define OFF_U10  6212      // 100
#define OFF_ROT  6312      // 64 (c[16], s[16], p[16] ints, q[16] ints)
#define OFF_MISC 6376      // y3[4], vec[3]

// block-shared weights (float offsets, after the wave arenas)
#define W_W1P 0            // 60*32 = 1920 (zero padded W1, [k][n])
#define W_W2  1920         // 20*10
#define W_W3  2120         // 10*2
#define W_WL  2140         // 17*3
#define W_BL  2191         // 17
#define W_TOT 2208

#define SMEM_F (WPB * ARENA_F + W_TOT)   // 15008 floats = 60032 bytes

// ---------------------------------------------------------------------------
// Wave-parallel cyclic Jacobi with tournament (round-robin) pairing.
// ---------------------------------------------------------------------------
__device__ __forceinline__ void jacobi_inplace(float* A, float* U, float* rot,
                                               int n, int sweeps, int lane) {
  const int npairs = n >> 1;
  const int rounds = n - 1;
  float* rc = rot;
  float* rs = rot + 16;
  int*   rp = (int*)(rot + 32);
  int*   rq = (int*)(rot + 48);
  for (int sw = 0; sw < sweeps; ++sw) {
    for (int r = 0; r < rounds; ++r) {
      if (lane < npairs) {
        int a = (r + lane) % (n - 1);
        int b = (lane == 0) ? (n - 1) : ((r + (n - 1) - lane) % (n - 1));
        int p = a < b ? a : b;
        int q = a < b ? b : a;
        float app = A[p * n + p];
        float aqq = A[q * n + q];
        float apq = A[p * n + q];
        float c = 1.0f, s = 0.0f;
        if (__builtin_fabsf(apq) > 1e-30f) {
          float tau = (aqq - app) / (2.0f * apq);
          float t = (tau >= 0.0f ? 1.0f : -1.0f) /
                    (__builtin_fabsf(tau) + __builtin_sqrtf(1.0f + tau * tau));
          c = 1.0f / __builtin_sqrtf(1.0f + t * t);
          s = t * c;
        }
        rc[lane] = c; rs[lane] = s; rp[lane] = p; rq[lane] = q;
      }
      __syncthreads();
      for (int idx = lane; idx < npairs * n; idx += 32) {   // row phase
        int t = idx / n, j = idx - t * n;
        int p = rp[t], q = rq[t];
        float c = rc[t], s = rs[t];
        float apj = A[p * n + j], aqj = A[q * n + j];
        A[p * n + j] = c * apj - s * aqj;
        A[q * n + j] = s * apj + c * aqj;
      }
      __syncthreads();
      for (int idx = lane; idx < npairs * n; idx += 32) {   // col + U phase
        int t = idx / n, i = idx - t * n;
        int p = rp[t], q = rq[t];
        float c = rc[t], s = rs[t];
        float aip = A[i * n + p], aiq = A[i * n + q];
        A[i * n + p] = c * aip - s * aiq;
        A[i * n + q] = s * aip + c * aiq;
        float uip = U[i * n + p], uiq = U[i * n + q];
        U[i * n + p] = c * uip - s * uiq;
        U[i * n + q] = s * uip + c * uiq;
      }
      __syncthreads();
    }
  }
}

// H = U * diag(max(diag(A), eps)) * U^T
__device__ __forceinline__ void rectify_to(const float* A, const float* U,
                                           float* H, int n, int lane) {
  for (int idx = lane; idx < n * n; idx += 32) {
    int i = idx / n, j = idx - (idx / n) * n;
    float acc = 0.0f;
    for (int k = 0; k < n; ++k) {
      float lam = fmaxf(A[k * n + k], RECT_EPS);
      acc = __builtin_fmaf(U[i * n + k] * lam, U[j * n + k], acc);
    }
    H[idx] = acc;
  }
  __syncthreads();
}

// Masked LDS read of the compact X tile (zero outside 57x57, NaN-safe).
__device__ __forceinline__ float ldx(const float* xp, int row, int k) {
  return (row < DD && k < DD) ? xp[row * DD + k] : 0.0f;
}

__global__ __launch_bounds__(THREADS) void spdnet_fused_kernel(
    const float* __restrict__ x,   // [B,57,57]
    const float* __restrict__ W1,  // [57,20]
    const float* __restrict__ W2,  // [20,10]
    const float* __restrict__ W3,  // [10,2]
    const float* __restrict__ Wl,  // [17,3]
    const float* __restrict__ bl,  // [17]
    float* __restrict__ out,       // [B,17]
    int Btot) {
  __shared__ float smem[SMEM_F];

  const int tid    = threadIdx.x;
  const int lane   = tid & 31;
  const int waveId = tid >> 5;
  const int b      = blockIdx.x * WPB + waveId;
  const bool active = (b < Btot);
  const int bb     = active ? b : (Btot - 1);

  float* warena = smem + waveId * ARENA_F;
  float* wgt    = smem + WPB * ARENA_F;

  float* xp   = warena + OFF_XPAD;
  float* tp   = warena + OFF_TPAD;
  float* a20  = warena + OFF_A20;
  float* u20  = warena + OFF_U20;
  float* a10  = warena + OFF_A10;
  float* u10  = warena + OFF_U10;
  float* rot  = warena + OFF_ROT;
  float* y3   = warena + OFF_MISC;
  float* vecb = warena + OFF_MISC + 4;

  float* w1p = wgt + W_W1P;
  float* w2s = wgt + W_W2;
  float* w3s = wgt + W_W3;
  float* wls = wgt + W_WL;
  float* bls = wgt + W_BL;

  // ---- kick off TDM load of this wave's X tile: global -> LDS (TENSORcnt) ----
  const float* Xb = x + (size_t)bb * (DD * DD);
#if __has_builtin(__builtin_amdgcn_tensor_load_to_lds)
  {
    unsigned long long ga = (unsigned long long)(uintptr_t)Xb;
    unsigned int ldsb = (unsigned int)((waveId * ARENA_F + OFF_XPAD) * 4);
    v4u g0;
    g0[0] = 1u;                                  // count=1, user descriptor
    g0[1] = ldsb;                                // lds_addr (bytes)
    g0[2] = (unsigned int)(ga & 0xFFFFFFFFull);  // global_addr[31:0]
    g0[3] = (unsigned int)((ga >> 32) & 0x1FFFFFFull) | (2u << 30);  // [56:32], type=2
    v8i g1;
    g1[0] = (int)(2u << 16);          // workgroup_mask=0, data_size=4B
    g1[1] = (int)(57u << 16);         // tensor_dim0 = 57 (lo16 here)
    g1[2] = (int)(57u << 16);         // tensor_dim0 hi=0 | tensor_dim1 lo = 57
    g1[3] = (int)(57u << 16);         // tensor_dim1 hi=0 | tile_dim0 = 57
    g1[4] = (int)57u;                 // tile_dim1 = 57, tile_dim2 = 0
    g1[5] = (int)57u;                 // tensor_dim0_stride = 57
    g1[6] = 0;                        // stride hi | tensor_dim1_stride lo
    g1[7] = 0;                        // tensor_dim1_stride hi
    v4i gz = {0, 0, 0, 0};
#if defined(__clang_major__) && (__clang_major__ >= 23)
    v8i gz8 = {0, 0, 0, 0, 0, 0, 0, 0};
    __builtin_amdgcn_tensor_load_to_lds(g0, g1, gz, gz, gz8, 0);
#else
    __builtin_amdgcn_tensor_load_to_lds(g0, g1, gz, gz, 0);
#endif
  }
#else
  for (int i = lane; i < DD * DD; i += 32) xp[i] = Xb[i];
#endif

  // ---- stage weights into LDS while TDM runs (block-cooperative) ----
  for (int i = tid; i < 60 * NP; i += THREADS) {
    int k = i / NP, n = i - (i / NP) * NP;
    w1p[i] = (k < DD && n < 20) ? W1[k * 20 + n] : 0.0f;
  }
  for (int i = tid; i < 200; i += THREADS) w2s[i] = W2[i];
  for (int i = tid; i < 20;  i += THREADS) w3s[i] = W3[i];
  for (int i = tid; i < 51;  i += THREADS) wls[i] = Wl[i];
  for (int i = tid; i < 17;  i += THREADS) bls[i] = bl[i];

#if __has_builtin(__builtin_amdgcn_s_wait_tensorcnt)
  __builtin_amdgcn_s_wait_tensorcnt(0);
#endif
  __syncthreads();

  const int m   = lane & 15;   // A-operand row / B-operand col for this lane
  const int hlf = lane >> 4;   // lane half selects K sub-pair
  const int kA0 = hlf ? 2 : 0;
  const int kA1 = hlf ? 3 : 1;

  // ---- GEMM1: Tpad(64x32) = X(57x57, masked) * W1pad(60x32), fp32 WMMA ----
  for (int nt = 0; nt < 2; ++nt) {
    v2f bvv[15];                      // B tiles reused across all 4 M-tiles
#pragma unroll
    for (int kk = 0; kk < 15; ++kk) {
      const int k0 = kk * 4;
      bvv[kk].x = w1p[(k0 + kA0) * NP + nt * 16 + m];
      bvv[kk].y = w1p[(k0 + kA1) * NP + nt * 16 + m];
    }
    for (int mt = 0; mt < 4; ++mt) {
      const int row = mt * 16 + m;
      v8f acc = {0.f, 0.f, 0.f, 0.f, 0.f, 0.f, 0.f, 0.f};
#pragma unroll
      for (int kk = 0; kk < 15; ++kk) {
        const int k0 = kk * 4;
        v2f av;
        av.x = ldx(xp, row, k0 + kA0);
        av.y = ldx(xp, row, k0 + kA1);
        acc = __builtin_amdgcn_wmma_f32_16x16x4_f32(
            false, av, false, bvv[kk], (short)0, acc, false, false);
      }
#pragma unroll
      for (int r = 0; r < 8; ++r) {
        int trow = mt * 16 + r + hlf * 8;
        tp[trow * NP + nt * 16 + m] = acc[r];
      }
    }
  }
  __syncthreads();

  // ---- GEMM2: A20(20x20) = W1^T(32x60) * Tpad(60x32), fp32 WMMA ----
  for (int mt = 0; mt < 2; ++mt) {
    v2f avv[15];                      // (W1^T)[m][k] tiles reused across N-tiles
#pragma unroll
    for (int kk = 0; kk < 15; ++kk) {
      const int k0 = kk * 4;
      avv[kk].x = w1p[(k0 + kA0) * NP + mt * 16 + m];
      avv[kk].y = w1p[(k0 + kA1) * NP + mt * 16 + m];
    }
    for (int nt = 0; nt < 2; ++nt) {
      v8f acc = {0.f, 0.f, 0.f, 0.f, 0.f, 0.f, 0.f, 0.f};
#pragma unroll
      for (int kk = 0; kk < 15; ++kk) {
        const int k0 = kk * 4;
        v2f bv;
        bv.x = tp[(k0 + kA0) * NP + nt * 16 + m];
        bv.y = tp[(k0 + kA1) * NP + nt * 16 + m];
        acc = __builtin_amdgcn_wmma_f32_16x16x4_f32(
            false, avv[kk], false, bv, (short)0, acc, false, false);
      }
#pragma unroll
      for (int r = 0; r < 8; ++r) {
        int row = mt * 16 + r + hlf * 8;
        int col = nt * 16 + m;
        if (row < 20 && col < 20) a20[row * 20 + col] = acc[r];
      }
    }
  }
  // U20 = I
  for (int i = lane; i < 400; i += 32) {
    int r = i / 20, c = i - (i / 20) * 20;
    u20[i] = (r == c) ? 1.0f : 0.0f;
  }
  __syncthreads();

  // ---- eigh(20) + rectify -> H1 (reuse xp region) ----
  jacobi_inplace(a20, u20, rot, 20, 8, lane);
  float* h1 = xp;
  rectify_to(a20, u20, h1, 20, lane);

  // ---- transform 2: A10 = W2^T H1 W2 (VALU) ----
  float* t2 = tp;  // [20 x 10]
  for (int idx = lane; idx < 200; idx += 32) {
    int i = idx / 10, c = idx - (idx / 10) * 10;
    float acc = 0.0f;
    for (int j = 0; j < 20; ++j)
      acc = __builtin_fmaf(h1[i * 20 + j], w2s[j * 10 + c], acc);
    t2[idx] = acc;
  }
  __syncthreads();
  for (int idx = lane; idx < 100; idx += 32) {
    int a = idx / 10, c = idx - (idx / 10) * 10;
    float acc = 0.0f;
    for (int i = 0; i < 20; ++i)
      acc = __builtin_fmaf(w2s[i * 10 + a], t2[i * 10 + c], acc);
    a10[idx] = acc;
    u10[idx] = (a == c) ? 1.0f : 0.0f;
  }
  __syncthreads();

  // ---- eigh(10) + rectify -> H2 ----
  jacobi_inplace(a10, u10, rot, 10, 8, lane);
  float* h2 = xp + 512;
  rectify_to(a10, u10, h2, 10, lane);

  // ---- transform 3: Y3 = W3^T H2 W3 (2x2) ----
  float* t3 = tp + 256;  // [10 x 2]
  for (int idx = lane; idx < 20; idx += 32) {
    int i = idx >> 1, c = idx & 1;
    float acc = 0.0f;
    for (int j = 0; j < 10; ++j)
      acc = __builtin_fmaf(h2[i * 10 + j], w3s[j * 2 + c], acc);
    t3[idx] = acc;
  }
  __syncthreads();
  if (lane < 4) {
    int a = lane >> 1, c = lane & 1;
    float acc = 0.0f;
    for (int i = 0; i < 10; ++i)
      acc = __builtin_fmaf(w3s[i * 2 + a], t3[i * 2 + c], acc);
    y3[lane] = acc;
  }
  __syncthreads();

  // ---- 2x2 rectify + log map + triu vectorization (closed form) ----
  if (lane == 0) {
    float a = y3[0], bq = y3[1], c = y3[3];
    float mid  = 0.5f * (a + c);
    float dif  = 0.5f * (a - c);
    float disc = __builtin_sqrtf(dif * dif + bq * bq);
    float s1 = mid + disc, s2 = mid - disc;
    float l1 = fmaxf(s1, RECT_EPS), l2 = fmaxf(s2, RECT_EPS);
    float g1 = __builtin_logf(l1), g2 = __builtin_logf(l2);
    float v1, v2;
    if (__builtin_fabsf(bq) > 1e-30f) {
      v1 = bq; v2 = s1 - a;
      float inv = 1.0f / __builtin_sqrtf(v1 * v1 + v2 * v2);
      v1 *= inv; v2 *= inv;
    } else if (a >= c) { v1 = 1.0f; v2 = 0.0f; }
    else               { v1 = 0.0f; v2 = 1.0f; }
    vecb[0] = g1 * v1 * v1 + g2 * v2 * v2;
    vecb[1] = SQRT2_F * (g1 - g2) * v1 * v2;
    vecb[2] = g1 * v2 * v2 + g2 * v1 * v1;
  }
  __syncthreads();

  // ---- linear head: out = vec @ Wl^T + bl ----
  if (active && lane < 17) {
    float r = bls[lane];
    r = __builtin_fmaf(wls[lane * 3 + 0], vecb[0], r);
    r = __builtin_fmaf(wls[lane * 3 + 1], vecb[1], r);
    r = __builtin_fmaf(wls[lane * 3 + 2], vecb[2], r);
    out[(size_t)bb * 17 + lane] = r;
  }
}

extern "C" void kernel_launch(void* const* d_in, const int* in_sizes, int n_in,
                              void* d_out, int out_size, void* d_ws, size_t ws_size,
                              hipStream_t stream) {
  const float* x  = (const float*)d_in[0];
  const float* W1 = (const float*)d_in[1];
  const float* W2 = (const float*)d_in[2];
  const float* W3 = (const float*)d_in[3];
  const float* Wl = (const float*)d_in[4];
  const float* bl = (const float*)d_in[5];
  float* out = (float*)d_out;

  const int Btot = in_sizes[0] / (DD * DD);
  const int grid = (Btot + WPB - 1) / WPB;
  spdnet_fused_kernel<<<grid, THREADS, 0, stream>>>(x, W1, W2, W3, Wl, bl, out, Btot);
}